// QuantumDisorderedFCGraph_27324581937504
// MI455X (gfx1250) — compile-verified
//
#include <hip/hip_runtime.h>
#include <hip/hip_bf16.h>

// MI455X / gfx1250 implementation of the batched quantum-circuit forward pass.
// One 256-thread block per batch row; one wave32 per index-register row.
// Each row's 256 complex amplitudes are a 16x16 complex matrix in LDS; every
// gate layer is A * M * B^T with A,B = Kronecker products of 2x2 unitaries,
// executed as chains of v_wmma_f32_16x16x4_f32.

typedef float v2f __attribute__((ext_vector_type(2)));
typedef float v8f __attribute__((ext_vector_type(8)));

struct cplx { float re, im; };

__device__ __forceinline__ cplx cmul(cplx a, cplx b) {
    return { a.re * b.re - a.im * b.im, a.re * b.im + a.im * b.re };
}

// PennyLane qml.Rot(phi, theta, omega) = RZ(omega) RY(theta) RZ(phi)
__device__ __forceinline__ void rot_gate(float phi, float th, float om, cplx U[2][2]) {
    float c = __cosf(0.5f * th);
    float s = __sinf(0.5f * th);
    float a = -0.5f * (phi + om);   // ep = exp(-i(phi+om)/2)
    float b =  0.5f * (phi - om);   // em = exp(+i(phi-om)/2)
    float epr = __cosf(a), epi = __sinf(a);
    float emr = __cosf(b), emi = __sinf(b);
    U[0][0] = {  epr * c,  epi * c };   // ep * c
    U[0][1] = { -emr * s, -emi * s };   // -em * s
    U[1][0] = {  emr * s, -emi * s };   // conj(em) * s
    U[1][1] = {  epr * c, -epi * c };   // conj(ep) * c
}

// Entry [row][col] of U[0] (x) U[1] (x) U[2] (x) U[3]  (U[0] = MSB qubit)
__device__ __forceinline__ cplx kron4(const cplx U[4][2][2], int row, int col) {
    cplx e = U[0][(row >> 3) & 1][(col >> 3) & 1];
    e = cmul(e, U[1][(row >> 2) & 1][(col >> 2) & 1]);
    e = cmul(e, U[2][(row >> 1) & 1][(col >> 1) & 1]);
    e = cmul(e, U[3][ row       & 1][ col       & 1]);
    return e;
}

#define WMMA4(A, B, C) \
    __builtin_amdgcn_wmma_f32_16x16x4_f32(false, (A), false, (B), (short)0, (C), false, false)

// LEFT:  M' = G * M      (G = kron of gates on qubits 0..3)
// !LEFT: M' = M * G^T    (G = kron of gates on qubits 4..7)
// Srow: 256 interleaved complex (16x16 row-major). Wave-private region.
template <bool LEFT>
__device__ __forceinline__ void apply_group(float2* Srow, int lane, const cplx U[4][2][2]) {
    const int hb = lane >> 4;     // half-wave selector
    const int lo = lane & 15;
    v8f Dr = {}; v8f Di = {};
    #pragma unroll
    for (int k = 0; k < 4; ++k) {
        v2f mr, mi, gr, gi, gin;
        #pragma unroll
        for (int v = 0; v < 2; ++v) {
            int rk = 4 * k + v + 2 * hb;          // K-index within full 16
            // M operand: B-layout chunk for LEFT (M[rk][lo]); A-layout for RIGHT (M[lo][rk])
            float2 m = LEFT ? Srow[rk * 16 + lo] : Srow[lo * 16 + rk];
            mr[v] = m.x; mi[v] = m.y;
            // Gate operand: A-layout of G (LEFT) and B-layout of G^T (RIGHT) share
            // the same per-lane indexing formula G[lo][rk].
            cplx g = kron4(U, lo, rk);
            gr[v] = g.re; gi[v] = g.im; gin[v] = -g.im;
        }
        if (LEFT) {
            Dr = WMMA4(gr,  mr, Dr);  Dr = WMMA4(gin, mi, Dr);
            Di = WMMA4(gr,  mi, Di);  Di = WMMA4(gi,  mr, Di);
        } else {
            Dr = WMMA4(mr, gr,  Dr);  Dr = WMMA4(mi, gin, Dr);
            Di = WMMA4(mr, gi,  Di);  Di = WMMA4(mi, gr,  Di);
        }
    }
    __asm__ __volatile__("" ::: "memory");  // keep all LDS loads before stores
    #pragma unroll
    for (int v = 0; v < 8; ++v) {
        int n = (v + 8 * hb) * 16 + lo;     // C/D layout -> matrix position
        Srow[n] = make_float2(Dr[v], Di[v]);
    }
}

// Forward basis map of the CNOT ring: for w=0..7, bit[(w+rng)%8] ^= bit[w]
// (qubit w lives at bit position 7-w of n).
__device__ __forceinline__ int cnot_map(int n, int rng) {
    int b[8];
    #pragma unroll
    for (int w = 0; w < 8; ++w) b[w] = (n >> (7 - w)) & 1;
    #pragma unroll
    for (int w = 0; w < 8; ++w) b[(w + rng) & 7] ^= b[w];
    int m = 0;
    #pragma unroll
    for (int w = 0; w < 8; ++w) m |= b[w] << (7 - w);
    return m;
}

__global__ __launch_bounds__(256)
void qc_wmma_kernel(const float* __restrict__ x, const float* __restrict__ wts,
                    float* __restrict__ out) {
    __shared__ float2 S[8 * 256];   // 8 rows x (16x16 complex)
    __shared__ float  gbuf[256];

    const int batch = blockIdx.x;
    const int tid   = threadIdx.x;
    const int row   = tid >> 5;     // wave id = index-register row
    const int lane  = tid & 31;
    const int hb    = lane >> 4;
    const int lo    = lane & 15;
    float2* Srow = &S[row * 256];

    // --- init |0...0>: row 0, n = 0 gets amplitude 1 ---
    #pragma unroll
    for (int v = 0; v < 8; ++v) {
        int n = (v + 8 * hb) * 16 + lo;
        Srow[n] = make_float2((row == 0 && n == 0) ? 1.0f : 0.0f, 0.0f);
    }

    // --- per-row encoding unitary V (faithful repro of CTRL_ROWS bug) ---
    // row0:{U0} row1:{U4 U2 U1} row3:{U6 U3} row5:{U5} row7:{U7} rows 2,4,6: identity
    int pl[3]; int np = 0;
    switch (row) {
        case 0: pl[0] = 0; np = 1; break;
        case 1: pl[0] = 1; pl[1] = 2; pl[2] = 4; np = 3; break;
        case 3: pl[0] = 3; pl[1] = 6; np = 2; break;
        case 5: pl[0] = 5; np = 1; break;
        case 7: pl[0] = 7; np = 1; break;
        default: np = 0; break;
    }
    cplx V[2][2] = { { {1.f,0.f}, {0.f,0.f} }, { {0.f,0.f}, {1.f,0.f} } };
    for (int i = 0; i < np; ++i) {
        int j = pl[i];
        cplx U[2][2];
        rot_gate(x[batch * 24 + j * 3 + 0],
                 x[batch * 24 + j * 3 + 1],
                 x[batch * 24 + j * 3 + 2], U);
        cplx Vn[2][2];
        #pragma unroll
        for (int a = 0; a < 2; ++a)
            #pragma unroll
            for (int b = 0; b < 2; ++b) {
                cplx t = cmul(U[a][0], V[0][b]);
                cplx u2 = cmul(U[a][1], V[1][b]);
                Vn[a][b] = { t.re + u2.re, t.im + u2.im };
            }
        V[0][0] = Vn[0][0]; V[0][1] = Vn[0][1];
        V[1][0] = Vn[1][0]; V[1][1] = Vn[1][1];
    }

    cplx G4[4][2][2];

    for (int rep = 0; rep < 2; ++rep) {
        // --- controlled encoding: V on every nn qubit of this row ---
        if (np > 0) {
            #pragma unroll
            for (int t = 0; t < 4; ++t) {
                G4[t][0][0] = V[0][0]; G4[t][0][1] = V[0][1];
                G4[t][1][0] = V[1][0]; G4[t][1][1] = V[1][1];
            }
            apply_group<true >(Srow, lane, G4);   // qubits 0..3
            apply_group<false>(Srow, lane, G4);   // qubits 4..7
        }
        // --- StronglyEntanglingLayers ---
        for (int l = 0; l < 2; ++l) {
            const float* wl = wts + ((rep * 2 + l) * 8) * 3;
            #pragma unroll
            for (int q = 0; q < 4; ++q)
                rot_gate(wl[q * 3 + 0], wl[q * 3 + 1], wl[q * 3 + 2], G4[q]);
            apply_group<true>(Srow, lane, G4);
            #pragma unroll
            for (int q = 0; q < 4; ++q)
                rot_gate(wl[(q + 4) * 3 + 0], wl[(q + 4) * 3 + 1], wl[(q + 4) * 3 + 2], G4[q]);
            apply_group<false>(Srow, lane, G4);

            // CNOT ring, range = l+1: scatter new[F(n)] = old[n]
            const int rng = l + 1;
            float2 vals[8]; int dst[8];
            #pragma unroll
            for (int v = 0; v < 8; ++v) {
                int n = (v + 8 * hb) * 16 + lo;
                vals[v] = Srow[n];
                dst[v]  = cnot_map(n, rng);
            }
            __asm__ __volatile__("" ::: "memory");
            #pragma unroll
            for (int v = 0; v < 8; ++v) Srow[dst[v]] = vals[v];
        }
    }

    __syncthreads();

    // --- readout: G[n] = sum_m w_m sum_j Re(conj(S[j,n]) S[j^m,n]),
    //     mask multiplicities: {0:1, 1:3, 3:2, 5:1, 7:1} ---
    {
        const int n = tid;
        float2 c[8];
        #pragma unroll
        for (int j = 0; j < 8; ++j) c[j] = S[j * 256 + n];
        float g0 = 0.f, g1 = 0.f, g3 = 0.f, g5 = 0.f, g7 = 0.f;
        #pragma unroll
        for (int j = 0; j < 8; ++j) {
            g0 += c[j].x * c[j].x + c[j].y * c[j].y;
            g1 += c[j].x * c[j ^ 1].x + c[j].y * c[j ^ 1].y;
            g3 += c[j].x * c[j ^ 3].x + c[j].y * c[j ^ 3].y;
            g5 += c[j].x * c[j ^ 5].x + c[j].y * c[j ^ 5].y;
            g7 += c[j].x * c[j ^ 7].x + c[j].y * c[j ^ 7].y;
        }
        gbuf[n] = g0 + 3.f * g1 + 2.f * g3 + g5 + g7;
    }
    __syncthreads();

    // --- out[q] = sum_n (1 - 2*bit_{7-q}(n)) * G[n]; wave `row` handles q = row ---
    {
        const int q = row;
        float acc = 0.f;
        #pragma unroll
        for (int i = 0; i < 8; ++i) {
            int n = lane + 32 * i;
            float s = ((n >> (7 - q)) & 1) ? -1.0f : 1.0f;
            acc += s * gbuf[n];
        }
        #pragma unroll
        for (int off = 16; off > 0; off >>= 1)
            acc += __shfl_xor(acc, off, 32);
        if (lane == 0) out[batch * 8 + q] = acc;
    }
}

extern "C" void kernel_launch(void* const* d_in, const int* in_sizes, int n_in,
                              void* d_out, int out_size, void* d_ws, size_t ws_size,
                              hipStream_t stream) {
    const float* x   = (const float*)d_in[0];   // (512, 24) float32
    const float* wts = (const float*)d_in[1];   // (2, 2, 8, 3) float32
    float* out = (float*)d_out;                 // (512, 8) float32
    const int batch = in_sizes[0] / 24;         // 512
    qc_wmma_kernel<<<batch, 256, 0, stream>>>(x, wts, out);
}